// EPcont_39470749450740
// MI455X (gfx1250) — compile-verified
//
#include <hip/hip_runtime.h>

// Problem constants (from reference)
#define B_ROWS   8192
#define NIN      784
#define NH       2048
#define NOUT     10
#define T_STEPS  15
#define DT       0.5f

// Tiling
#define MT        16     // batch rows per block
#define NTHREADS  512    // 16 waves of 32
#define NWAVES    (NTHREADS / 32)
#define TILES_PER_WAVE ((NH / 16) / NWAVES)   // 8
#define COLS_PER_LANE  (NH / 32)              // 64

// LDS strides. sA/sC padded against bank conflicts (788%64=20, 2052%64=4).
// sW0 is deliberately UNPADDED (2048) so the TDM copy is a flat 1-D transfer;
// dynamics reads sW0 with consecutive lanes -> consecutive banks, so row
// stride is irrelevant for conflicts there.
#define SA_STRIDE 788
#define SC_STRIDE 2052
#define SW_STRIDE 2048

typedef __attribute__((ext_vector_type(2))) float v2f;
typedef __attribute__((ext_vector_type(4))) float v4f;
typedef __attribute__((ext_vector_type(8))) float v8f;
typedef __attribute__((ext_vector_type(4))) unsigned int u32x4;
typedef __attribute__((ext_vector_type(8))) int i32x8;
typedef __attribute__((ext_vector_type(4))) int i32x4;

__device__ __forceinline__ float rho_f(float x) {
  return fminf(fmaxf(x, 0.0f), 1.0f);   // hard-sigmoid; also == clamp(x,0,1)
}

extern "C" __global__ __launch_bounds__(NTHREADS)
void epcont_fused(const float* __restrict__ data,
                  const float* __restrict__ s0_in,
                  const float* __restrict__ s1_in,
                  const float* __restrict__ W0,
                  const float* __restrict__ b0,
                  const float* __restrict__ W2,
                  const float* __restrict__ b2,
                  float* __restrict__ out)
{
  extern __shared__ float smem[];
  float* sA  = smem;                     // [MT][SA_STRIDE] rho(data) tile
  float* sC  = sA + MT * SA_STRIDE;      // [MT][SC_STRIDE] C = rho(data)@W2^T + b2 (loop-invariant)
  float* sW0 = sC + MT * SC_STRIDE;      // [NOUT][SW_STRIDE] raw W0 copy (via TDM)

  const int tid  = threadIdx.x;
  const int lane = tid & 31;
  const int wave = tid >> 5;
  const int r0   = blockIdx.x * MT;

  // ---------------- TDM: async-copy W0 (10x2048 f32, contiguous) into LDS ----------------
  // Issued once by wave 0 (TDM ignores EXEC; branch is wave-uniform so only
  // wave 0 issues). Other waves sync on the barrier below.
  if (tid < 32) {
#if __has_builtin(__builtin_amdgcn_tensor_load_to_lds)
    const unsigned lds_w0 = (unsigned)((const char*)sW0 - (const char*)smem);
    const unsigned long long ga = (unsigned long long)(uintptr_t)W0;
    const unsigned NW0 = NOUT * NH;             // 20480 f32 elements
    // D# group 0 (CDNA5 ISA)
    u32x4 g0;
    g0[0] = 1u;                                  // count=1, user mode, no gather
    g0[1] = lds_w0;                              // lds_addr (bytes)
    g0[2] = (unsigned)(ga & 0xffffffffu);        // global_addr[31:0]
    g0[3] = (unsigned)((ga >> 32) & 0x01ffffffu) // global_addr[56:32]
          | 0x80000000u;                         // type=2 ("image") in bits [127:126]
    // D# group 1 (§8.4): 1-D copy, data_size=4B, tile_dim0=tensor_dim0=20480
    i32x8 g1;
    g1[0] = (int)(2u << 16);                          // wg_mask=0 | data_size=2 (4B) | no flags
    g1[1] = (int)((NW0 & 0xffffu) << 16);             // abar_addr=0 | tensor_dim0[15:0]
    g1[2] = (int)((NW0 >> 16) & 0xffffu) | (1 << 16); // tensor_dim0[31:16] | tensor_dim1[15:0]=1
    g1[3] = (int)((NW0 & 0xffffu) << 16);             // tensor_dim1[31:16]=0 | tile_dim0=20480
    g1[4] = 0;                                        // tile_dim1=0 (unused), tile_dim2=0
    g1[5] = (int)NW0;                                 // tensor_dim0_stride[31:0]
    g1[6] = 0;                                        // stride0 hi | dim1_stride lo
    g1[7] = 0;                                        // dim1_stride hi
    i32x4 gz4 = {0, 0, 0, 0};                         // groups 2/3 unused (<=2-D tensor)
    i32x8 gz8 = {0, 0, 0, 0, 0, 0, 0, 0};            // extra group (6-arg toolchain form)
    __builtin_amdgcn_tensor_load_to_lds(g0, g1, gz4, gz4, gz8, 0);
    __builtin_amdgcn_s_wait_tensorcnt(0);
#else
    for (int idx = lane; idx < NOUT * NH; idx += 32) sW0[idx] = W0[idx];
#endif
  }

  // ---------------- stage rho(data) tile into LDS (elementwise transform) ----------------
  for (int idx = tid; idx < MT * NIN; idx += NTHREADS) {
    int row = idx / NIN;
    int k   = idx - row * NIN;
    sA[row * SA_STRIDE + k] = rho_f(data[(size_t)(r0 + row) * NIN + k]);
  }
  __syncthreads();

  // ---------------- phase 1: C tile via V_WMMA_F32_16X16X4_F32 ----------------
  // C[m,n] = sum_k rho(data)[m,k] * W2[n,k] + b2[n]; wave w owns N cols
  // [w*128, w*128+128) as 8 tiles of 16, K swept 4 at a time (196 WMMAs/tile).
  {
    const int hi = lane >> 4;           // 0: lanes 0-15, 1: lanes 16-31
    const int lm = lane & 15;
    // A 16x4 f32 layout: VGPR0 = {K0 (lo lanes), K2 (hi lanes)}, VGPR1 = {K1, K3}
    const int aoff = lm * SA_STRIDE + 2 * hi;

    for (int t = 0; t < TILES_PER_WAVE; ++t) {
      const int n0  = (wave * TILES_PER_WAVE + t) * 16;
      const int col = n0 + lm;                       // this lane's N column
      const float* wrow = W2 + (size_t)col * NIN;    // W2 row = B-matrix column
      __builtin_prefetch(wrow, 0, 0);                // global_prefetch_b8

      v8f acc = {};
      #pragma unroll 4
      for (int k0 = 0; k0 < NIN; k0 += 4) {
        v2f a = *(const v2f*)(sA + aoff + k0);       // ds_load_b64, 8B aligned
        v4f w4 = *(const v4f*)(wrow + k0);           // global_load_b128, 16B aligned
        // B 4x16 f32 layout mirrored from A: VGPR0 = {K0 lo, K2 hi}, VGPR1 = {K1 lo, K3 hi}
        v2f b;
        b.x = hi ? w4.z : w4.x;
        b.y = hi ? w4.w : w4.y;
        acc = __builtin_amdgcn_wmma_f32_16x16x4_f32(
            /*neg_a=*/false, a, /*neg_b=*/false, b,
            /*c_mod=*/(short)0, acc, /*reuse_a=*/false, /*reuse_b=*/false);
      }
      // C/D layout: VGPR e -> M = e + 8*hi, N = col. Fold in b2 here.
      const float bias  = b2[col];
      const int   mbase = hi * 8;
      #pragma unroll
      for (int e = 0; e < 8; ++e)
        sC[(mbase + e) * SC_STRIDE + col] = acc[e] + bias;
    }
  }
  __syncthreads();

  // ---------------- phase 2: T relaxation steps, one wave per batch row ----------------
  const int r = wave;      // row within tile
  const int c = lane;      // column phase; this lane owns cols j = c + 32*i

  float s1l[COLS_PER_LANE];
  {
    const float* s1row = s1_in + (size_t)(r0 + r) * NH;
    #pragma unroll
    for (int i = 0; i < COLS_PER_LANE; ++i) s1l[i] = s1row[c + 32 * i];
  }
  float s0l[NOUT];
  {
    const float* s0row = s0_in + (size_t)(r0 + r) * NOUT;
    #pragma unroll
    for (int o = 0; o < NOUT; ++o) s0l[o] = s0row[o];
  }
  const float* cRow = sC + r * SC_STRIDE;

  for (int tstep = 0; tstep < T_STEPS; ++tstep) {
    float a_[NOUT], p_[NOUT];
    #pragma unroll
    for (int o = 0; o < NOUT; ++o) { a_[o] = rho_f(s0l[o]); p_[o] = 0.0f; }

    #pragma unroll
    for (int i = 0; i < COLS_PER_LANE; ++i) {
      const int j = c + 32 * i;                 // lanes hit consecutive LDS banks
      float s   = s1l[i];
      float rs  = rho_f(s);
      float pre = cRow[j];                      // includes b2
      #pragma unroll
      for (int o = 0; o < NOUT; ++o) {
        float w = sW0[o * SW_STRIDE + j];       // one read serves both products
        p_[o] = fmaf(w, rs, p_[o]);             // W0 . rho(s1)  (for ds0)
        pre   = fmaf(a_[o], w, pre);            // rho(s0) . W0  (for ds1)
      }
      float rp = (s >= 0.0f && s <= 1.0f) ? 1.0f : 0.0f;   // rhop
      float ns = s + DT * (-s + rp * pre);
      s1l[i] = rho_f(ns);                       // clamp(.,0,1)
    }

    // reduce p_ over the 32 lanes of this wave (full row sum), then update s0.
    // b0[o] is wave-uniform w/ constant index -> scalar loads, hoisted to SGPRs.
    #pragma unroll
    for (int o = 0; o < NOUT; ++o) {
      float v = p_[o];
      v += __shfl_xor(v, 1, 32);
      v += __shfl_xor(v, 2, 32);
      v += __shfl_xor(v, 4, 32);
      v += __shfl_xor(v, 8, 32);
      v += __shfl_xor(v, 16, 32);
      float ns = s0l[o] + DT * (-s0l[o] + v + b0[o]);
      s0l[o] = rho_f(ns);
    }
  }

  // ---------------- write output: concat([s0, s1], axis=1) ----------------
  float* orow = out + (size_t)(r0 + r) * (NOUT + NH);
  if (c == 0) {
    #pragma unroll
    for (int o = 0; o < NOUT; ++o) orow[o] = s0l[o];
  }
  #pragma unroll
  for (int i = 0; i < COLS_PER_LANE; ++i) orow[NOUT + c + 32 * i] = s1l[i];
}

extern "C" void kernel_launch(void* const* d_in, const int* in_sizes, int n_in,
                              void* d_out, int out_size, void* d_ws, size_t ws_size,
                              hipStream_t stream)
{
  const float* data = (const float*)d_in[0];
  const float* s0   = (const float*)d_in[1];
  const float* s1   = (const float*)d_in[2];
  const float* W0   = (const float*)d_in[3];
  const float* b0   = (const float*)d_in[4];
  const float* W2   = (const float*)d_in[5];
  const float* b2   = (const float*)d_in[6];
  float* out = (float*)d_out;
  (void)in_sizes; (void)n_in; (void)out_size; (void)d_ws; (void)ws_size;

  const size_t lds_bytes =
      (size_t)(MT * SA_STRIDE + MT * SC_STRIDE + NOUT * SW_STRIDE) * sizeof(float); // ~258 KB (<=320 KB/WGP)
  (void)hipFuncSetAttribute((const void*)epcont_fused,
                            hipFuncAttributeMaxDynamicSharedMemorySize,
                            (int)lds_bytes);

  dim3 grid(B_ROWS / MT);   // 512 blocks
  dim3 block(NTHREADS);     // 512 threads = 16 waves
  epcont_fused<<<grid, block, lds_bytes, stream>>>(data, s0, s1, W0, b0, W2, b2, out);
}